// GIN_Model_90769838833738
// MI455X (gfx1250) — compile-verified
//
#include <hip/hip_runtime.h>

#define D 128
#define NLAYERS 4

typedef __attribute__((ext_vector_type(2))) float v2f;
typedef __attribute__((ext_vector_type(8))) float v8f;

// ---------------------------------------------------------------- utilities

__global__ void zero_f32_kernel(float* __restrict__ p, long n) {
    long i = (long)blockIdx.x * blockDim.x + threadIdx.x;
    if (i < n) p[i] = 0.0f;
}

// One thread per edge: fp32 degree accumulation (native global_atomic_add_f32).
__global__ void degree_kernel(const int* __restrict__ src, const int* __restrict__ dst,
                              float* __restrict__ deg_src, float* __restrict__ deg_dst,
                              int E) {
    int e = blockIdx.x * blockDim.x + threadIdx.x;
    if (e < E) {
        unsafeAtomicAdd(&deg_src[src[e]], 1.0f);
        unsafeAtomicAdd(&deg_dst[dst[e]], 1.0f);
    }
}

// In-place: deg -> clip(deg,1)^-0.5   (norm_src and norm_dst are contiguous,
// so one launch covers both).
__global__ void norm_kernel(float* __restrict__ deg, long n) {
    long i = (long)blockIdx.x * blockDim.x + threadIdx.x;
    if (i < n) deg[i] = rsqrtf(fmaxf(deg[i], 1.0f));
}

// ------------------------------------------------------- SpMM (gather+scatter)
// 32 threads per edge; each thread moves a float4 feature chunk.
// agg[dst] += h[src] * norm_src[src]
__global__ void spmm_scatter_kernel(const float* __restrict__ hin,
                                    const float* __restrict__ norm_src,
                                    const int* __restrict__ src,
                                    const int* __restrict__ dst,
                                    float* __restrict__ agg, int E) {
    int t = blockIdx.x * blockDim.x + threadIdx.x;
    int e = t >> 5;              // edge index
    int q = (t & 31) << 2;       // feature offset (0,4,...,124)
    if (e >= E) return;
    int s = src[e];
    int d = dst[e];
    float4 v = *(const float4*)(hin + (long)s * D + q);
    float ns = norm_src[s];
    float* outp = agg + (long)d * D + q;
    unsafeAtomicAdd(outp + 0, v.x * ns);
    unsafeAtomicAdd(outp + 1, v.y * ns);
    unsafeAtomicAdd(outp + 2, v.z * ns);
    unsafeAtomicAdd(outp + 3, v.w * ns);
}

// ------------------------------------------------------------ WMMA fp32 GEMM
// out[16 rows x 128 cols per block] = (agg * norm_dst) @ W + bias
// 8 waves/block; wave w computes the 16x16 tile at columns [16w, 16w+16).
// Uses V_WMMA_F32_16X16X4_F32, K accumulated over 128 in steps of 4.
__global__ void __launch_bounds__(256)
gcn_gemm_wmma_kernel(const float* __restrict__ agg,
                     const float* __restrict__ norm_dst,
                     const float* __restrict__ W,      // [D x D], row-major (k, n)
                     const float* __restrict__ bias,   // [D]
                     float* __restrict__ out, int N) {
    __shared__ float sA[16][D + 4];                    // +4 pad: conflict-free rows

    const int tid     = threadIdx.x;
    const int rowBase = blockIdx.x * 16;

    // Cooperative load of the 16x128 A tile, fused with the norm_dst scaling.
    for (int idx = tid; idx < 16 * D; idx += 256) {
        int r = idx >> 7;            // 0..15
        int c = idx & (D - 1);       // 0..127
        int row = rowBase + r;
        float v = 0.0f;
        if (row < N) v = agg[(long)row * D + c] * norm_dst[row];
        sA[r][c] = v;
    }
    __syncthreads();

    const int wave  = tid >> 5;
    const int lane  = tid & 31;
    const int laneM = lane & 15;            // A row this lane holds
    const int koff  = (lane >> 4) << 1;     // lanes 0-15: K+0/1, lanes 16-31: K+2/3
    const int n     = (wave << 4) + laneM;  // output column

    // C/D layout: VGPR v, lanes 0-15 -> (M=v, N=lane); lanes 16-31 -> (M=v+8).
    // Bias depends only on N, so every accumulator element seeds with bias[n].
    v8f c;
    const float bv = bias[n];
#pragma unroll
    for (int v = 0; v < 8; ++v) c[v] = bv;

#pragma unroll
    for (int k0 = 0; k0 < D; k0 += 4) {
        v2f a, b;
        // A 16x4 fragment (ISA 7.12.2: VGPR0=K+koff, VGPR1=K+koff+1)
        a.x = sA[laneM][k0 + koff];
        a.y = sA[laneM][k0 + koff + 1];
        // B 4x16 fragment, mirrored K pairing; row-major W: W[k][n]
        b.x = W[(long)(k0 + koff) * D + n];
        b.y = W[(long)(k0 + koff + 1) * D + n];
        c = __builtin_amdgcn_wmma_f32_16x16x4_f32(
                /*neg_a=*/false, a, /*neg_b=*/false, b,
                /*c_mod=*/(short)0, c, /*reuse_a=*/false, /*reuse_b=*/false);
    }

    const int mBase = rowBase + ((lane >> 4) << 3);
#pragma unroll
    for (int v = 0; v < 8; ++v) {
        int row = mBase + v;
        if (row < N) out[(long)row * D + n] = c[v];
    }
}

// ------------------------------------------------------------------- launcher

extern "C" void kernel_launch(void* const* d_in, const int* in_sizes, int n_in,
                              void* d_out, int out_size, void* d_ws, size_t ws_size,
                              hipStream_t stream) {
    const float* feat = (const float*)d_in[0];   // [N, D]
    const float* W    = (const float*)d_in[1];   // [4, D, D]
    const float* bias = (const float*)d_in[2];   // [4, D]
    const int*   src  = (const int*)d_in[3];     // [E]
    const int*   dst  = (const int*)d_in[4];     // [E]

    const int N = in_sizes[0] / D;
    const int E = in_sizes[3];

    // Workspace layout (floats): norm_src[N] | norm_dst[N] | agg[N*D] | hbuf[N*D]
    float* ws       = (float*)d_ws;
    float* norm_src = ws;
    float* norm_dst = ws + N;
    float* agg      = ws + 2L * N;
    float* hbuf     = agg + (long)N * D;
    float* outf     = (float*)d_out;

    const long twoN = 2L * N;
    const long ND   = (long)N * D;

    // 1) degrees -> norms (re-initialized every call: deterministic)
    zero_f32_kernel<<<(int)((twoN + 255) / 256), 256, 0, stream>>>(norm_src, twoN);
    degree_kernel<<<(E + 255) / 256, 256, 0, stream>>>(src, dst, norm_src, norm_dst, E);
    norm_kernel<<<(int)((twoN + 255) / 256), 256, 0, stream>>>(norm_src, twoN);

    // 2) layers; ping-pong feat -> hbuf -> d_out -> hbuf -> d_out
    const long edgeThreads = (long)E * 32;
    const int  edgeGrid    = (int)((edgeThreads + 255) / 256);
    const int  gemmGrid    = (N + 15) / 16;

    const float* hin = feat;
    for (int layer = 0; layer < NLAYERS; ++layer) {
        float* hout = (layer & 1) ? outf : hbuf;   // layer 3 (last) -> d_out

        zero_f32_kernel<<<(int)((ND + 255) / 256), 256, 0, stream>>>(agg, ND);
        spmm_scatter_kernel<<<edgeGrid, 256, 0, stream>>>(hin, norm_src, src, dst, agg, E);
        gcn_gemm_wmma_kernel<<<gemmGrid, 256, 0, stream>>>(
            agg, norm_dst, W + (long)layer * D * D, bias + (long)layer * D, hout, N);

        hin = hout;
    }
}